// MMNL_loss_37168646980397
// MI455X (gfx1250) — compile-verified
//
#include <hip/hip_runtime.h>
#include <hip/hip_bf16.h>
#include <stdint.h>

#define MM   8
#define BB   65536
#define CC   100
#define NB   32                 // customers per block
#define NBLK (BB / NB)          // 2048 blocks
#define NTHR 288                // 9 wave32s: 8 z-waves + 1 x/y wave
#define ROWB (CC * 4)           // 400 bytes per row
#define CHUNKB (NB * ROWB)      // 12800 bytes per chunk
#define CHUNK16 (CHUNKB / 16)   // 800 16-byte pieces per chunk
#define NCHUNK 10               // 8 z + x + y

// CDNA5 async global -> LDS copy, 16 bytes per lane (gfx1250: tracked by ASYNCcnt)
__device__ __forceinline__ void async_copy16(uint32_t lds_byte_addr, const void* gaddr) {
    asm volatile("global_load_async_to_lds_b128 %0, %1, off"
                 :: "v"(lds_byte_addr), "v"(gaddr)
                 : "memory");
}

__device__ __forceinline__ void wait_async_all() {
    asm volatile("s_wait_asynccnt 0" ::: "memory");
}

__global__ __launch_bounds__(NTHR)
void mmnl_partial_kernel(const float* __restrict__ x,
                         const float* __restrict__ y,
                         const float* __restrict__ z,
                         const float* __restrict__ alpha,
                         const float* __restrict__ u_prev,
                         const float* __restrict__ u_scalar,
                         float* __restrict__ partials) {
    __shared__ __align__(16) char smem[NCHUNK * CHUNKB];   // 128000 B staged tile
    __shared__ float ezS[MM][NB];   // per (m, customer) sum of exp(z)
    __shared__ float sxS[NB];       // per customer sum of exp(x)
    __shared__ float xyS[NB];       // per customer x . y
    __shared__ int   chS[NB];       // choice index, -1 if no purchase

    const int tid = threadIdx.x;
    const int b0  = blockIdx.x * NB;

    const uint32_t smem_base = (uint32_t)(uintptr_t)(void*)&smem[0];

    // ---- Stage 1: async copy 10 contiguous chunks into LDS (coalesced b128).
    // Outer loop fully unrolled -> chunk is a compile-time constant, so the
    // global base resolves statically and the inner loop is uniform.
    #pragma unroll
    for (int chunk = 0; chunk < NCHUNK; ++chunk) {
        const char* gbase;
        if (chunk < MM) {
            gbase = (const char*)(z + (size_t)chunk * BB * CC + (size_t)b0 * CC);
        } else if (chunk == MM) {
            gbase = (const char*)(x + (size_t)b0 * CC);
        } else {
            gbase = (const char*)(y + (size_t)b0 * CC);
        }
        for (int p = tid; p < CHUNK16; p += NTHR) {
            async_copy16(smem_base + (uint32_t)(chunk * CHUNKB + p * 16),
                         gbase + (size_t)p * 16);
        }
    }
    wait_async_all();
    __syncthreads();

    // ---- Stage 2: serial per-thread row reductions out of LDS ----
    const int r = tid >> 5;      // wave id 0..8
    const int j = tid & 31;      // customer within block

    if (r < MM) {
        // wave r reduces z rows of model m=r for all 32 customers (one per lane)
        const float4* row = (const float4*)(smem + ((size_t)r * NB + j) * ROWB);
        float s = 0.0f;
        #pragma unroll 5
        for (int c4 = 0; c4 < CC / 4; ++c4) {
            float4 v = row[c4];
            s += __expf(v.x) + __expf(v.y) + __expf(v.z) + __expf(v.w);
        }
        ezS[r][j] = s;
    } else {
        // wave 8: x row (exp-sum, x.y) and y row (choice detection)
        const float4* xrow = (const float4*)(smem + ((size_t)MM * NB + j) * ROWB);
        const float4* yrow = (const float4*)(smem + ((size_t)(MM + 1) * NB + j) * ROWB);
        float sx = 0.0f, xy = 0.0f;
        int ch = -1;
        #pragma unroll 5
        for (int c4 = 0; c4 < CC / 4; ++c4) {
            float4 xv = xrow[c4];
            float4 yv = yrow[c4];
            sx += __expf(xv.x) + __expf(xv.y) + __expf(xv.z) + __expf(xv.w);
            xy += xv.x * yv.x + xv.y * yv.y + xv.z * yv.z + xv.w * yv.w;
            if (yv.x > 0.5f) ch = 4 * c4 + 0;
            if (yv.y > 0.5f) ch = 4 * c4 + 1;
            if (yv.z > 0.5f) ch = 4 * c4 + 2;
            if (yv.w > 0.5f) ch = 4 * c4 + 3;
        }
        sxS[j] = sx;
        xyS[j] = xy;
        chS[j] = ch;
    }
    __syncthreads();

    // ---- Stage 3: wave 8 combines mixture terms and reduces the block ----
    if (r == MM) {
        const int   ch    = chS[j];
        const bool  has_y = (ch >= 0);
        const float eu    = __expf(u_scalar[0]);

        const float* zrows = (const float*)smem;   // chunk m at m*NB*CC floats
        float g = 0.0f;
        #pragma unroll
        for (int m = 0; m < MM; ++m) {
            const float eup   = __expf(u_prev[m]);
            const float denom = eup + ezS[m][j];
            float num;
            if (has_y) {
                num = __expf(zrows[((size_t)m * NB + j) * CC + ch]);
            } else {
                num = eup;
            }
            g += alpha[m] * num / denom;
        }
        const float num_x   = has_y ? __expf(xyS[j]) : eu;
        const float denom_x = eu + sxS[j];
        float contrib = num_x / (denom_x * g);

        // wave32 butterfly reduction over the 32 customers
        #pragma unroll
        for (int off = 16; off > 0; off >>= 1)
            contrib += __shfl_xor(contrib, off, 32);
        if (j == 0)
            partials[blockIdx.x] = contrib;
    }
}

__global__ __launch_bounds__(256)
void mmnl_finalize_kernel(const float* __restrict__ partials,
                          float* __restrict__ out) {
    __shared__ float red[256];
    float s = 0.0f;
    for (int i = threadIdx.x; i < NBLK; i += 256)
        s += partials[i];
    red[threadIdx.x] = s;
    __syncthreads();
    for (int stride = 128; stride > 0; stride >>= 1) {
        if (threadIdx.x < stride)
            red[threadIdx.x] += red[threadIdx.x + stride];
        __syncthreads();
    }
    if (threadIdx.x == 0)
        out[0] = -red[0] / (float)BB;
}

extern "C" void kernel_launch(void* const* d_in, const int* in_sizes, int n_in,
                              void* d_out, int out_size, void* d_ws, size_t ws_size,
                              hipStream_t stream) {
    (void)in_sizes; (void)n_in; (void)out_size; (void)ws_size;
    const float* x      = (const float*)d_in[0];
    const float* y      = (const float*)d_in[1];
    const float* z      = (const float*)d_in[2];
    const float* alpha  = (const float*)d_in[3];
    const float* u_prev = (const float*)d_in[4];
    const float* u      = (const float*)d_in[5];
    float* out      = (float*)d_out;
    float* partials = (float*)d_ws;   // NBLK floats = 8 KB

    mmnl_partial_kernel<<<dim3(NBLK), dim3(NTHR), 0, stream>>>(
        x, y, z, alpha, u_prev, u, partials);
    mmnl_finalize_kernel<<<dim3(1), dim3(256), 0, stream>>>(partials, out);
}